// RGCNGatherMM_3908420239950
// MI455X (gfx1250) — compile-verified
//
#include <hip/hip_runtime.h>

typedef __attribute__((ext_vector_type(2))) float v2f;
typedef __attribute__((ext_vector_type(8))) float v8f;

#define NRELS 16
#define DIM 64
#define BLOCKS_PER_REL 160

// ws int layout:
//   [0..15]    counts
//   [16..32]   offsets (17)
//   [33..48]   cursors
//   [64..64+E) sorted edge ids

__global__ void rgcn_zero_kernel(float* __restrict__ out, size_t n, int* __restrict__ wsHdr) {
    size_t i = (size_t)blockIdx.x * blockDim.x + threadIdx.x;
    size_t stride = (size_t)gridDim.x * blockDim.x;
    for (; i < n; i += stride) out[i] = 0.0f;
    if (blockIdx.x == 0 && threadIdx.x < 64) wsHdr[threadIdx.x] = 0;
}

__global__ void rgcn_count_kernel(const int* __restrict__ et, int* __restrict__ counts, int E) {
    __shared__ int h[NRELS];
    if (threadIdx.x < NRELS) h[threadIdx.x] = 0;
    __syncthreads();
    for (int i = blockIdx.x * blockDim.x + threadIdx.x; i < E; i += gridDim.x * blockDim.x)
        atomicAdd(&h[et[i]], 1);
    __syncthreads();
    if (threadIdx.x < NRELS) atomicAdd(&counts[threadIdx.x], h[threadIdx.x]);
}

__global__ void rgcn_scan_kernel(int* __restrict__ ws) {
    if (threadIdx.x == 0) {
        ws[16] = 0;
        for (int r = 0; r < NRELS; ++r) ws[17 + r] = ws[16 + r] + ws[r];
        for (int r = 0; r < NRELS; ++r) ws[33 + r] = ws[16 + r];
    }
}

__global__ void rgcn_scatter_kernel(const int* __restrict__ et, int* __restrict__ cursors,
                                    int* __restrict__ sorted, int E) {
    for (int i = blockIdx.x * blockDim.x + threadIdx.x; i < E; i += gridDim.x * blockDim.x) {
        int r = et[i];
        int pos = atomicAdd(&cursors[r], 1);
        sorted[pos] = i;
    }
}

// Grid: (BLOCKS_PER_REL, NRELS). 4 waves/block; wave w owns output columns [16w,16w+16).
// Each wave holds its relation's full 64x16 B slice in registers across all tiles.
// Partial tiles: invalid rows are clamped to a duplicate valid row (D row m depends only
// on A row m, and invalid D rows are never scattered), so no masking in the hot loop.
__global__ __launch_bounds__(128) void rgcn_gemm_kernel(
    const float* __restrict__ feat, const float* __restrict__ weight,
    const int* __restrict__ src, const int* __restrict__ dst,
    const int* __restrict__ sorted, const int* __restrict__ offsets,
    float* __restrict__ out) {
    __shared__ int sRow[16];  // source-node row per A row
    __shared__ int sDst[16];  // destination-node row per D row

    int rel = blockIdx.y;
    int start = offsets[rel];
    int cnt = offsets[rel + 1] - start;
    int numTiles = (cnt + 15) >> 4;
    if (numTiles == 0) return;

    int t = threadIdx.x;
    int lane = t & 31;
    int wave = t >> 5;
    int half = lane >> 4;   // splits the K range per ISA fragment layout
    int mn   = lane & 15;   // A: row M; B/D: column N

    // Load this wave's B slice (W_rel columns [16w,16w+16)) into 32 VGPRs, once.
    const float* bPtr = weight + (size_t)rel * DIM * DIM + (wave * 16 + mn);
    v2f bfrag[16];
#pragma unroll
    for (int kc = 0; kc < 16; ++kc) {
        int ka = 4 * kc + 2 * half;
        bfrag[kc].x = bPtr[(size_t)ka * DIM];
        bfrag[kc].y = bPtr[(size_t)(ka + 1) * DIM];
    }

    int colBase = wave * 16;

    for (int tile = blockIdx.x; tile < numTiles; tile += gridDim.x) {
        int rowBase = tile << 4;
        bool full = (rowBase + 16) <= cnt;   // block-uniform
        if (t < 16) {
            int idx = rowBase + t;
            if (idx >= cnt) idx = cnt - 1;   // clamp: duplicate row, output discarded
            int e = sorted[start + idx];
            sRow[t] = src[e];
            sDst[t] = dst[e];
        }
        __syncthreads();

        const float* aPtr = feat + (size_t)sRow[mn] * DIM;  // global address space
        v8f c = {};
#pragma unroll
        for (int kc = 0; kc < 16; ++kc) {
            int ka = 4 * kc + 2 * half;
            v2f a = *(const v2f*)(aPtr + ka);   // A[m][ka], A[m][ka+1]
            c = __builtin_amdgcn_wmma_f32_16x16x4_f32(
                /*neg_a=*/false, a, /*neg_b=*/false, bfrag[kc],
                /*c_mod=*/(short)0, c, /*reuse_a=*/false, /*reuse_b=*/false);
        }

        // D layout: VGPR j holds row (j + 8*half), column mn. Scatter-add to out[dst].
        if (full) {
#pragma unroll
            for (int j = 0; j < 8; ++j) {
                int m = j + 8 * half;
                atomicAdd(&out[(size_t)sDst[m] * DIM + colBase + mn], c[j]);
            }
        } else {
            int rem = cnt - rowBase;
#pragma unroll
            for (int j = 0; j < 8; ++j) {
                int m = j + 8 * half;
                if (m < rem) {
                    atomicAdd(&out[(size_t)sDst[m] * DIM + colBase + mn], c[j]);
                }
            }
        }
        __syncthreads();  // protect LDS before next tile's staging
    }
}

// Fallback (only if d_ws is too small for the sort buffer): one edge per block.
__global__ void rgcn_fallback_kernel(const float* __restrict__ feat, const float* __restrict__ w,
                                     const int* __restrict__ src, const int* __restrict__ dst,
                                     const int* __restrict__ et, float* __restrict__ out, int E) {
    int e = blockIdx.x;
    if (e >= E) return;
    int col = threadIdx.x;
    const float* h = feat + (size_t)src[e] * DIM;
    const float* wr = w + (size_t)et[e] * DIM * DIM + col;
    float acc = 0.0f;
#pragma unroll
    for (int k = 0; k < DIM; ++k) acc += h[k] * wr[(size_t)k * DIM];
    atomicAdd(&out[(size_t)dst[e] * DIM + col], acc);
}

extern "C" void kernel_launch(void* const* d_in, const int* in_sizes, int n_in,
                              void* d_out, int out_size, void* d_ws, size_t ws_size,
                              hipStream_t stream) {
    const float* feat   = (const float*)d_in[0];
    const float* weight = (const float*)d_in[1];
    const int*   src    = (const int*)d_in[2];
    const int*   dst    = (const int*)d_in[3];
    const int*   et     = (const int*)d_in[4];
    float* out = (float*)d_out;
    int E = in_sizes[2];

    int* ws_i = (int*)d_ws;
    size_t need = (64 + (size_t)E) * sizeof(int);

    rgcn_zero_kernel<<<1024, 256, 0, stream>>>(out, (size_t)out_size, ws_i);

    if (ws_size >= need) {
        rgcn_count_kernel<<<512, 256, 0, stream>>>(et, ws_i, E);
        rgcn_scan_kernel<<<1, 32, 0, stream>>>(ws_i);
        rgcn_scatter_kernel<<<512, 256, 0, stream>>>(et, ws_i + 33, ws_i + 64, E);
        dim3 grid(BLOCKS_PER_REL, NRELS);
        rgcn_gemm_kernel<<<grid, 128, 0, stream>>>(feat, weight, src, dst,
                                                   ws_i + 64, ws_i + 16, out);
    } else {
        rgcn_fallback_kernel<<<E, DIM, 0, stream>>>(feat, weight, src, dst, et, out, E);
    }
}